// VolumeRenderingNerf_29480655520026
// MI455X (gfx1250) — compile-verified
//
#include <hip/hip_runtime.h>
#include <stdint.h>

#define EPS 1e-7f
#define WAVES_PER_BLOCK 8
#define NBUF 3   // depth-2 async pipeline

// ---- address-space helpers -------------------------------------------------
typedef __attribute__((address_space(1))) int GINT;   // global int
typedef __attribute__((address_space(3))) int LINT;   // LDS int

__device__ __forceinline__ GINT* to_global_i32(const void* p) {
    // global virtual addresses are numerically identical to flat addresses
    return (GINT*)(uintptr_t)p;
}
__device__ __forceinline__ LINT* to_lds_i32(void* p) {
    // flat -> LDS addrspacecast (strips the aperture)
    return (LINT*)p;
}

// ---- CDNA5 async global->LDS copy -----------------------------------------
__device__ __forceinline__ void async_load_b32(const float* g, float* l) {
#if __has_builtin(__builtin_amdgcn_global_load_async_to_lds_b32)
    __builtin_amdgcn_global_load_async_to_lds_b32(to_global_i32(g), to_lds_i32(l), 0, 0);
#else
    unsigned lds_off = (unsigned)(uintptr_t)to_lds_i32(l);
    uint64_t gaddr   = (uint64_t)(uintptr_t)g;
    asm volatile("global_load_async_to_lds_b32 %0, %1, off"
                 :: "v"(lds_off), "v"(gaddr) : "memory");
#endif
}

template <int N>
__device__ __forceinline__ void wait_asynccnt() {
#if __has_builtin(__builtin_amdgcn_s_wait_asynccnt)
    __builtin_amdgcn_s_wait_asynccnt(N);
#else
    asm volatile("s_wait_asynccnt %0" :: "i"(N) : "memory");
#endif
    asm volatile("" ::: "memory");
}

__device__ __forceinline__ void wait_dscnt0() {
    asm volatile("s_wait_dscnt 0" ::: "memory");
}

// Issue one 32-sample chunk (5 coalesced async b32 loads, EXEC kept full via
// address clamping so ASYNCcnt stays deterministic).
__device__ __forceinline__ void issue_chunk(const float* __restrict__ sig,
                                            const float* __restrict__ dt,
                                            const float* __restrict__ rgb,
                                            int base, int end, int lane,
                                            float* buf /* 160 floats */) {
    int gi = base + lane;            // sigma / dt sample index
    int gmax = end - 1;
    if (gi > gmax) gi = gmax;
    async_load_b32(sig + gi, buf + lane);
    async_load_b32(dt  + gi, buf + 32 + lane);

    int f0   = base * 3 + lane;      // rgb flat float index, 96 contiguous
    int fmax = end * 3 - 1;
    int fa = f0;       if (fa > fmax) fa = fmax;
    int fb = f0 + 32;  if (fb > fmax) fb = fmax;
    int fc = f0 + 64;  if (fc > fmax) fc = fmax;
    async_load_b32(rgb + fa, buf + 64 + lane);
    async_load_b32(rgb + fb, buf + 96 + lane);
    async_load_b32(rgb + fc, buf + 128 + lane);
}

__global__ __launch_bounds__(256) void
VolumeRenderingNerf_kernel(const float* __restrict__ sigma,
                           const float* __restrict__ dt,
                           const float* __restrict__ rgb,
                           const int*   __restrict__ ray_start,
                           float*       __restrict__ out,
                           int R, int Ntot) {
    // per-wave triple-buffered staging: [sigma 32 | dt 32 | rgb 96] floats
    __shared__ float stage[WAVES_PER_BLOCK][NBUF][160];

    const int lane = threadIdx.x & 31;
    const int wid  = (int)threadIdx.x >> 5;
    const int ray  = (int)blockIdx.x * WAVES_PER_BLOCK + wid;
    if (ray >= R) return;  // wave-uniform

    const int start = ray_start[ray];
    const int end   = (ray + 1 < R) ? ray_start[ray + 1] : Ntot;
    const int nch   = (end - start + 31) >> 5;

    float carry = 0.f;                 // running sum of log(a) (log transmittance)
    float wsum = 0.f, cr = 0.f, cg = 0.f, cb = 0.f;

    // prologue: fill pipeline with chunks 0 and 1
    issue_chunk(sigma, dt, rgb, start, end, lane, &stage[wid][0][0]);
    if (nch > 1)
        issue_chunk(sigma, dt, rgb, start + 32, end, lane, &stage[wid][1][0]);

    int bufIdx = 0;
    for (int c = 0; c < nch; ++c) {
        const int base = start + (c << 5);
        float* buf = &stage[wid][bufIdx][0];

        if (c + 2 < nch) {
            // buffer (c+2)%3 last held chunk c-1, consumed last iteration;
            // drain its LDS reads before the async engine overwrites it.
            wait_dscnt0();
            int nb = bufIdx + 2; if (nb >= NBUF) nb -= NBUF;
            issue_chunk(sigma, dt, rgb, base + 64, end, lane, &stage[wid][nb][0]);
            wait_asynccnt<10>();  // chunks c+1, c+2 may remain outstanding
        } else if (c + 1 < nch) {
            wait_asynccnt<5>();   // chunk c+1 may remain outstanding
        } else {
            wait_asynccnt<0>();
        }

        const float s = buf[lane];
        const float d = buf[32 + lane];
        const float r = buf[64 + 3 * lane + 0];   // stride-3 dwords: bank-conflict free
        const float g = buf[64 + 3 * lane + 1];
        const float b = buf[64 + 3 * lane + 2];

        const bool valid = (base + lane) < end;
        const float alpha = valid ? (1.0f - __expf(-s * d)) : 0.0f;
        const float la    = valid ? __logf(1.0f - alpha + EPS) : 0.0f;

        // wave32 inclusive scan of log(a)
        float inc = la;
        #pragma unroll
        for (int off = 1; off < 32; off <<= 1) {
            float t = __shfl_up(inc, off, 32);
            if (lane >= off) inc += t;
        }
        const float excl = inc - la;               // exclusive prefix
        const float T    = __expf(carry + excl);   // transmittance
        const float w    = alpha * T;

        wsum += w;
        cr += w * r;  cg += w * g;  cb += w * b;
        carry += __shfl(inc, 31, 32);              // chunk total -> carry

        if (++bufIdx >= NBUF) bufIdx = 0;
    }

    // wave reduction of the 4 accumulators
    #pragma unroll
    for (int off = 16; off >= 1; off >>= 1) {
        wsum += __shfl_xor(wsum, off, 32);
        cr   += __shfl_xor(cr,   off, 32);
        cg   += __shfl_xor(cg,   off, 32);
        cb   += __shfl_xor(cb,   off, 32);
    }

    if (lane == 0) {
        float* o = out + (size_t)ray * 5;
        o[0] = cr;
        o[1] = cg;
        o[2] = cb;
        o[3] = wsum;
        o[4] = __expf(carry);   // bg transmittance
    }
}

extern "C" void kernel_launch(void* const* d_in, const int* in_sizes, int n_in,
                              void* d_out, int out_size, void* d_ws, size_t ws_size,
                              hipStream_t stream) {
    const float* sigma     = (const float*)d_in[0];  // [N,1]
    const float* dt        = (const float*)d_in[1];  // [N,1]
    const float* rgb       = (const float*)d_in[2];  // [N,3]
    // d_in[3] = ray_id (not needed; segment bounds come from ray_start)
    const int*   ray_start = (const int*)d_in[4];    // [R]

    const int Ntot = in_sizes[0];
    const int R    = in_sizes[4];
    float* out = (float*)d_out;

    const int blocks = (R + WAVES_PER_BLOCK - 1) / WAVES_PER_BLOCK;
    hipLaunchKernelGGL(VolumeRenderingNerf_kernel,
                       dim3(blocks), dim3(WAVES_PER_BLOCK * 32), 0, stream,
                       sigma, dt, rgb, ray_start, out, R, Ntot);
}